// RadiusInteractionGraph_43198781063281
// MI455X (gfx1250) — compile-verified
//
#include <hip/hip_runtime.h>
#include <hip/hip_bf16.h>

#define KNBR 32
#define CAP 64
#define WAVES 4
#define CUT2 25.0f
#define BIGF 1e30f

typedef __attribute__((ext_vector_type(2))) float v2f;
typedef __attribute__((ext_vector_type(8))) float v8f;

__global__ __launch_bounds__(32 * WAVES)
void radius_graph_wmma(const float* __restrict__ pos,
                       const int* __restrict__ batch,
                       float* __restrict__ outRow,
                       float* __restrict__ outCol,
                       float* __restrict__ outW,
                       float* __restrict__ outM,
                       int n) {
  __shared__ float s_d2[WAVES][16 * CAP];
  __shared__ int   s_id[WAVES][16 * CAP];
  __shared__ int   s_cnt[WAVES][16];
  __shared__ float s_tsq[WAVES][16];
  __shared__ int   s_tb[WAVES][16];

  const int lane = threadIdx.x & 31;
  const int wv   = threadIdx.x >> 5;
  const int tile = blockIdx.x * WAVES + wv;
  const int i0   = tile * 16;
  if (i0 >= n) return;                 // wave-uniform exit (EXEC stays all-1s for WMMA)

  float* d2b  = &s_d2[wv][0];
  int*   idb  = &s_id[wv][0];
  int*   cntb = &s_cnt[wv][0];

  const int half = lane >> 4;          // 0: lanes 0-15, 1: lanes 16-31
  const int l16  = lane & 15;

  // ---- load this lane's target (each half loads the same 16 targets) ----
  const int   tm = i0 + l16;
  const float tx = pos[3 * tm + 0];
  const float ty = pos[3 * tm + 1];
  const float tz = pos[3 * tm + 2];
  const int   tbat = batch[tm];

  if (lane < 16) {
    s_tsq[wv][lane] = tx * tx + ty * ty + tz * tz;
    s_tb[wv][lane]  = tbat;
    cntb[lane]      = 0;
  }
  asm volatile("s_wait_dscnt 0" ::: "memory");

  // ---- A fragment: 16x4 f32 (rows = targets, K = {x,y,z,0}) ----
  // ISA layout: lanes 0-15 hold K0,K1 ; lanes 16-31 hold K2,K3 of row (lane&15)
  v2f a;
  a.x = half ? tz : tx;
  a.y = half ? 0.0f : ty;

  // per-accumulator-row target stats (row v -> target v or v+8 by lane half)
  float tq[8];
  int   tbq[8];
#pragma unroll
  for (int v = 0; v < 8; ++v) {
    const int mr = v + (half ? 8 : 0);
    tq[v]  = s_tsq[wv][mr];
    tbq[v] = s_tb[wv][mr];
  }

  // ---- candidate span for this tile: sorted batch -> binary search ----
  const int g0 = batch[i0];
  const int g1 = batch[i0 + 15];
  int lo = 0;
  {
    int c = n;
    while (c > 0) {
      int st = c >> 1, mid = lo + st;
      if (batch[mid] < g0) { lo = mid + 1; c -= st + 1; } else c = st;
    }
  }
  int hi = 0;
  {
    int c = n;
    while (c > 0) {
      int st = c >> 1, mid = hi + st;
      if (batch[mid] <= g1) { hi = mid + 1; c -= st + 1; } else c = st;
    }
  }

  // ---- sweep candidate tiles: one V_WMMA_F32_16X16X4_F32 per 16x16 tile ----
  for (int j0 = (lo & ~15); j0 < hi; j0 += 16) {
    __builtin_prefetch(&pos[3 * (j0 + 16)], 0, 1);   // global_prefetch_b8

    const int   jn = j0 + l16;
    const int   jc = (jn < n) ? jn : (n - 1);
    const float cx = pos[3 * jc + 0];
    const float cy = pos[3 * jc + 1];
    const float cz = pos[3 * jc + 2];
    const int   cb = batch[jc];
    const float csq = cx * cx + cy * cy + cz * cz;

    // B fragment: 4x16 f32, columns = candidates
    v2f b;
    b.x = half ? cz : cx;
    b.y = half ? 0.0f : cy;

    v8f cacc = {};
    cacc = __builtin_amdgcn_wmma_f32_16x16x4_f32(
        /*neg_a=*/false, a, /*neg_b=*/false, b,
        /*c_mod=*/(short)0, cacc, /*reuse_a=*/false, /*reuse_b=*/false);

    // D row v (+8 in upper half) = target, column (lane&15) = candidate
#pragma unroll
    for (int v = 0; v < 8; ++v) {
      const int mr = v + (half ? 8 : 0);
      const int mg = i0 + mr;
      float d2 = tq[v] + csq - 2.0f * cacc[v];
      d2 = d2 < 0.0f ? 0.0f : d2;
      const bool ok = (jn < hi) && (jn != mg) && (cb == tbq[v]) && (d2 <= CUT2);
      if (ok) {
        const int slot = atomicAdd(&cntb[mr], 1);   // ds_add_rtn_u32
        if (slot < CAP) {
          d2b[mr * CAP + slot] = d2;
          idb[mr * CAP + slot] = jn;
        }
      }
    }
  }
  asm volatile("s_wait_dscnt 0" ::: "memory");

  // ---- per-target K-nearest selection (lanes 0-15, one target each) ----
  if (lane < 16) {
    const int t  = lane;
    int m = cntb[t];
    if (m > CAP) m = CAP;
    const int gi = i0 + t;
    const long long base = (long long)gi * KNBR;

    for (int k = 0; k < KNBR; ++k) {
      float bd = BIGF;
      int   bs = -1;
      for (int s = 0; s < m; ++s) {
        const float dv = d2b[t * CAP + s];
        if (dv < bd) { bd = dv; bs = s; }
      }
      float w, msk;
      int   nbr;
      if (bs >= 0) {
        nbr = idb[t * CAP + bs];
        w   = sqrtf(bd);
        msk = 1.0f;
        d2b[t * CAP + bs] = BIGF;   // remove from pool
      } else {
        nbr = gi;                   // pad: self-loop, weight 0, mask 0
        w   = 0.0f;
        msk = 0.0f;
      }
      outRow[base + k] = (float)nbr;
      outCol[base + k] = (float)gi;
      outW[base + k]   = w;
      outM[base + k]   = msk;
    }
  }
}

extern "C" void kernel_launch(void* const* d_in, const int* in_sizes, int n_in,
                              void* d_out, int out_size, void* d_ws, size_t ws_size,
                              hipStream_t stream) {
  const float* pos   = (const float*)d_in[0];   // [N,3] float32
  const int*   batch = (const int*)d_in[1];     // [N]   int32 (sorted)
  const int    n     = in_sizes[1];             // N

  float* out = (float*)d_out;                   // row ++ col ++ weight ++ mask
  const size_t NK = (size_t)n * KNBR;
  float* outRow = out;
  float* outCol = out + NK;
  float* outW   = out + 2 * NK;
  float* outM   = out + 3 * NK;

  const int tiles  = (n + 15) / 16;
  const int blocks = (tiles + WAVES - 1) / WAVES;
  hipLaunchKernelGGL(radius_graph_wmma, dim3(blocks), dim3(32 * WAVES), 0, stream,
                     pos, batch, outRow, outCol, outW, outM, n);
}